// CrossAttnYZ_53446573031919
// MI455X (gfx1250) — compile-verified
//
#include <hip/hip_runtime.h>

// ---------------------------------------------------------------------------
// CrossAttnYZ on MI455X (gfx1250, wave32, WMMA)
// sim = ([Q|Qm] [K|Km]^T) * scale -> softmax -> *V -> Wo projection.
// fp32 WMMA (V_WMMA_F32_16X16X4_F32) GEMMs with double-buffered LDS tiles
// fed by GLOBAL_LOAD_ASYNC_TO_LDS_B128 (ASYNCcnt-pipelined) when available.
// ---------------------------------------------------------------------------

#define DIMX   512
#define NSEQ   2048
#define BATCH  4
#define HEADS  8
#define DHEAD  64
#define BHT    (BATCH * HEADS)   // 32 batch-heads
#define DKC    128               // concatenated [Q|Qm] head dim
#define SCL    0.125f            // 64^-0.5

typedef __attribute__((ext_vector_type(2))) float v2f;
typedef __attribute__((ext_vector_type(8))) float v8f;
typedef int v4i __attribute__((vector_size(16)));   // matches builtin param type

#define AS_LD 20                 // 16 + 4 pad: 80B rows (16B aligned, bank-spread)
#define BS_LD 68                 // 64 + 4 pad: 272B rows (16B aligned)
#define A_TILE (128 * AS_LD)
#define B_TILE (16 * BS_LD)

// ---- async global->LDS path (guarded; falls back to direct staging) -------
#if defined(__has_builtin)
#if __has_builtin(__builtin_amdgcn_global_load_async_to_lds_b128)
#define HAS_ASYNC_LDS 1
#endif
#endif
#ifndef HAS_ASYNC_LDS
#define HAS_ASYNC_LDS 0
#endif

__device__ __forceinline__ void cp_b128(const float* __restrict__ g, float* l)
{
#if HAS_ASYNC_LDS
  __builtin_amdgcn_global_load_async_to_lds_b128(
      (__attribute__((address_space(1))) v4i*)g,
      (__attribute__((address_space(3))) v4i*)l, 0, 0);
#else
  *(float4*)l = *(const float4*)g;
#endif
}

template <int N>
__device__ __forceinline__ void wait_async()
{
#if HAS_ASYNC_LDS
#if __has_builtin(__builtin_amdgcn_s_wait_asynccnt)
  __builtin_amdgcn_s_wait_asynccnt(N);
#else
  asm volatile("s_wait_asynccnt %0" ::"i"(N) : "memory");
#endif
#endif
}

// Stage one 128x16 A tile + one 16x64 B tile into an LDS buffer.
// Per thread: 2 + 1 b128 copies => 3 ASYNCcnt increments per wave per stage.
__device__ __forceinline__ void stage_tile(const float* __restrict__ abase, int lda,
                                           const float* __restrict__ bbase, int ldb,
                                           float* As, float* Bs, int tid)
{
#pragma unroll
  for (int i = 0; i < 2; ++i) {
    const int f = tid + i * 256;
    const int row = f >> 2, c4 = (f & 3) * 4;
    cp_b128(abase + (size_t)row * lda + c4, As + row * AS_LD + c4);
  }
  {
    const int row = tid >> 4, c4 = (tid & 15) * 4;
    cp_b128(bbase + (size_t)row * ldb + c4, Bs + row * BS_LD + c4);
  }
}

// Per-wave 32x32 micro-tile over a BK=16 LDS tile:
// 4 k-steps x (2x2) of V_WMMA_F32_16X16X4_F32.
__device__ __forceinline__ void wmma_btile(const float* __restrict__ As,
                                           const float* __restrict__ Bs,
                                           v8f acc[2][2], int wr, int wc, int lane)
{
  const int g  = lane >> 4;   // half-wave selects K pair {0,1} vs {2,3}
  const int lm = lane & 15;
#pragma unroll
  for (int kk = 0; kk < 4; ++kk) {
    const int k2 = kk * 4 + g * 2;
    v2f a[2], b[2];
#pragma unroll
    for (int t = 0; t < 2; ++t) {
      const float* ap = As + (wr + t * 16 + lm) * AS_LD + k2;
      a[t].x = ap[0];
      a[t].y = ap[1];
      const int c = wc + t * 16 + lm;
      b[t].x = Bs[(k2 + 0) * BS_LD + c];
      b[t].y = Bs[(k2 + 1) * BS_LD + c];
    }
#pragma unroll
    for (int tm = 0; tm < 2; ++tm)
#pragma unroll
      for (int tn = 0; tn < 2; ++tn)
        acc[tm][tn] = __builtin_amdgcn_wmma_f32_16x16x4_f32(
            false, a[tm], false, b[tn], (short)0, acc[tm][tn], false, false);
  }
}

__device__ __forceinline__ void zero_acc(v8f acc[2][2])
{
#pragma unroll
  for (int i = 0; i < 2; ++i)
#pragma unroll
    for (int j = 0; j < 2; ++j)
#pragma unroll
      for (int r = 0; r < 8; ++r) acc[i][j][r] = 0.0f;
}

// Double-buffered main loop shared by all GEMM kernels (caller supplies a
// lambda mapping k-tile index -> (abase, bbase)).
template <typename StageFn>
__device__ __forceinline__ void gemm_db(int nkt, StageFn&& stage,
                                        float* AsBuf, float* BsBuf,
                                        v8f acc[2][2], int wr, int wc, int lane)
{
  stage(0, 0);
  for (int kt = 0; kt < nkt; ++kt) {
    const int cur = kt & 1;
    if (kt + 1 < nkt) {
      stage(kt + 1, cur ^ 1);
      wait_async<3>();        // previous stage's 3 copies have landed
    } else {
      wait_async<0>();
    }
    __syncthreads();
    wmma_btile(AsBuf + cur * A_TILE, BsBuf + cur * B_TILE, acc, wr, wc, lane);
    __syncthreads();
  }
}

// ---------------------------------------------------------------------------
// Kernel 1: fused projections.  job z: 0:Q=x*Wq 1:Qm=met*Wq 2:K=yz*Wk
// 3:Km=met*Wk 4:V=yz*Wv.  Q/Qm -> Qc[bh][n][128] (met half in cols 64..127);
// K/Km -> KcT[bh][128][n] (TRANSPOSED so the QK B-tile is row-major);
// V -> Vh[bh][n][64].
// ---------------------------------------------------------------------------
__global__ __launch_bounds__(256) void proj_kernel(
    const float* __restrict__ x, const float* __restrict__ met,
    const float* __restrict__ yz,
    const float* __restrict__ Wq, const float* __restrict__ Wk,
    const float* __restrict__ Wv,
    float* __restrict__ Qc, float* __restrict__ KcT, float* __restrict__ Vh)
{
  __shared__ float AsBuf[2 * A_TILE];
  __shared__ float BsBuf[2 * B_TILE];

  const float* X; const float* W; float* dst; int coloff; bool trans; int dld;
  switch (blockIdx.z) {
    case 0:  X = x;   W = Wq; dst = Qc;  coloff = 0;  trans = false; dld = DKC;   break;
    case 1:  X = met; W = Wq; dst = Qc;  coloff = 64; trans = false; dld = DKC;   break;
    case 2:  X = yz;  W = Wk; dst = KcT; coloff = 0;  trans = true;  dld = DKC;   break;
    case 3:  X = met; W = Wk; dst = KcT; coloff = 64; trans = true;  dld = DKC;   break;
    default: X = yz;  W = Wv; dst = Vh;  coloff = 0;  trans = false; dld = DHEAD; break;
  }

  const int tid = threadIdx.x, lane = tid & 31, wid = tid >> 5;
  const int wr = (wid >> 1) * 32, wc = (wid & 1) * 32;
  const int m0 = blockIdx.y * 128;
  const int n0 = blockIdx.x * 64;

  v8f acc[2][2];
  zero_acc(acc);

  gemm_db(DIMX / 16,
          [&](int kt, int buf) {
            stage_tile(X + (size_t)m0 * DIMX + kt * 16, DIMX,
                       W + (size_t)(kt * 16) * DIMX + n0, DIMX,
                       AsBuf + buf * A_TILE, BsBuf + buf * B_TILE, tid);
          },
          AsBuf, BsBuf, acc, wr, wc, lane);

  const int g = lane >> 4, lm = lane & 15;
#pragma unroll
  for (int tm = 0; tm < 2; ++tm)
#pragma unroll
    for (int tn = 0; tn < 2; ++tn) {
      const int j = n0 + wc + tn * 16 + lm;
      const int h = j >> 6, d = j & 63;
#pragma unroll
      for (int r = 0; r < 8; ++r) {
        const int m = m0 + wr + tm * 16 + g * 8 + r;
        const int b = m >> 11, n = m & (NSEQ - 1);
        const int bh = (b << 3) + h;
        const size_t idx = trans
            ? ((size_t)bh * DKC + coloff + d) * NSEQ + n
            : ((size_t)bh * NSEQ + n) * dld + coloff + d;
        dst[idx] = acc[tm][tn][r];
      }
    }
}

// ---------------------------------------------------------------------------
// Kernel 2: S = Qc * KcT * scale per batch-head (K-dim = 128) into attn.
// ---------------------------------------------------------------------------
__global__ __launch_bounds__(256) void qk_kernel(
    const float* __restrict__ Qc, const float* __restrict__ KcT,
    float* __restrict__ attn)
{
  __shared__ float AsBuf[2 * A_TILE];
  __shared__ float BsBuf[2 * B_TILE];

  const int bh = blockIdx.z;
  const float* Qb  = Qc  + (size_t)bh * NSEQ * DKC;
  const float* KbT = KcT + (size_t)bh * DKC * NSEQ;
  float* Sb = attn + (size_t)bh * NSEQ * NSEQ;

  const int tid = threadIdx.x, lane = tid & 31, wid = tid >> 5;
  const int wr = (wid >> 1) * 32, wc = (wid & 1) * 32;
  const int m0 = blockIdx.y * 128;   // query rows
  const int n0 = blockIdx.x * 64;    // key columns

  v8f acc[2][2];
  zero_acc(acc);

  gemm_db(DKC / 16,
          [&](int kt, int buf) {
            stage_tile(Qb + (size_t)m0 * DKC + kt * 16, DKC,
                       KbT + (size_t)(kt * 16) * NSEQ + n0, NSEQ,
                       AsBuf + buf * A_TILE, BsBuf + buf * B_TILE, tid);
          },
          AsBuf, BsBuf, acc, wr, wc, lane);

  const int g = lane >> 4, lm = lane & 15;
#pragma unroll
  for (int tm = 0; tm < 2; ++tm)
#pragma unroll
    for (int tn = 0; tn < 2; ++tn) {
      const int j = n0 + wc + tn * 16 + lm;
#pragma unroll
      for (int r = 0; r < 8; ++r) {
        const int m = m0 + wr + tm * 16 + g * 8 + r;
        Sb[(size_t)m * NSEQ + j] = acc[tm][tn][r] * SCL;
      }
    }
}

// ---------------------------------------------------------------------------
// Kernel 3: in-place row softmax over attn (one 256-thread block per row).
// ---------------------------------------------------------------------------
__global__ __launch_bounds__(256) void softmax_kernel(float* __restrict__ attn)
{
  float* p = attn + (size_t)blockIdx.x * NSEQ;
  __shared__ float red[8];

  float v[8];
  float mx = -__builtin_inff();
#pragma unroll
  for (int i = 0; i < 8; ++i) {
    v[i] = p[threadIdx.x + i * 256];
    mx = fmaxf(mx, v[i]);
  }
#pragma unroll
  for (int off = 16; off; off >>= 1) mx = fmaxf(mx, __shfl_xor(mx, off, 32));
  if ((threadIdx.x & 31) == 0) red[threadIdx.x >> 5] = mx;
  __syncthreads();
  float bm = red[0];
#pragma unroll
  for (int w = 1; w < 8; ++w) bm = fmaxf(bm, red[w]);

  float s = 0.0f;
#pragma unroll
  for (int i = 0; i < 8; ++i) {
    v[i] = __expf(v[i] - bm);
    s += v[i];
  }
#pragma unroll
  for (int off = 16; off; off >>= 1) s += __shfl_xor(s, off, 32);
  __syncthreads();
  if ((threadIdx.x & 31) == 0) red[threadIdx.x >> 5] = s;
  __syncthreads();
  float bs = 0.0f;
#pragma unroll
  for (int w = 0; w < 8; ++w) bs += red[w];
  const float inv = 1.0f / bs;
#pragma unroll
  for (int i = 0; i < 8; ++i) p[threadIdx.x + i * 256] = v[i] * inv;
}

// ---------------------------------------------------------------------------
// Kernel 4: Oh = attn * V per batch-head (M=2048, K=2048, N=64).
// ---------------------------------------------------------------------------
__global__ __launch_bounds__(256) void av_kernel(
    const float* __restrict__ attn, const float* __restrict__ Vh,
    float* __restrict__ Oh)
{
  __shared__ float AsBuf[2 * A_TILE];
  __shared__ float BsBuf[2 * B_TILE];

  const int bh = blockIdx.z;
  const float* Pb = attn + (size_t)bh * NSEQ * NSEQ;
  const float* Vb = Vh + (size_t)bh * NSEQ * DHEAD;
  float* Ob = Oh + (size_t)bh * NSEQ * DHEAD;

  const int tid = threadIdx.x, lane = tid & 31, wid = tid >> 5;
  const int wr = (wid >> 1) * 32, wc = (wid & 1) * 32;
  const int m0 = blockIdx.y * 128;

  v8f acc[2][2];
  zero_acc(acc);

  gemm_db(NSEQ / 16,
          [&](int kt, int buf) {
            stage_tile(Pb + (size_t)m0 * NSEQ + kt * 16, NSEQ,
                       Vb + (size_t)(kt * 16) * DHEAD, DHEAD,
                       AsBuf + buf * A_TILE, BsBuf + buf * B_TILE, tid);
          },
          AsBuf, BsBuf, acc, wr, wc, lane);

  const int g = lane >> 4, lm = lane & 15;
#pragma unroll
  for (int tm = 0; tm < 2; ++tm)
#pragma unroll
    for (int tn = 0; tn < 2; ++tn) {
      const int j = wc + tn * 16 + lm;
#pragma unroll
      for (int r = 0; r < 8; ++r) {
        const int m = m0 + wr + tm * 16 + g * 8 + r;
        Ob[(size_t)m * DHEAD + j] = acc[tm][tn][r];
      }
    }
}

// ---------------------------------------------------------------------------
// Kernel 5: out = regroup(Oh) * Wo + bo.  Each BK=16 k-slab stays inside one
// head, and all 128 rows of a block share one batch index, so the A base is
// affine per k-tile.
// ---------------------------------------------------------------------------
__global__ __launch_bounds__(256) void out_kernel(
    const float* __restrict__ Oh, const float* __restrict__ Wo,
    const float* __restrict__ bo, float* __restrict__ out)
{
  __shared__ float AsBuf[2 * A_TILE];
  __shared__ float BsBuf[2 * B_TILE];

  const int tid = threadIdx.x, lane = tid & 31, wid = tid >> 5;
  const int wr = (wid >> 1) * 32, wc = (wid & 1) * 32;
  const int m0 = blockIdx.y * 128;
  const int n0 = blockIdx.x * 64;
  const int b = m0 >> 11, nb = m0 & (NSEQ - 1);

  v8f acc[2][2];
  zero_acc(acc);

  gemm_db(DIMX / 16,
          [&](int kt, int buf) {
            const int k0 = kt * 16;
            const int h = k0 >> 6, dbase = k0 & 63;
            const float* abase =
                Oh + ((size_t)(((b << 3) + h) * NSEQ) + nb) * DHEAD + dbase;
            stage_tile(abase, DHEAD,
                       Wo + (size_t)k0 * DIMX + n0, DIMX,
                       AsBuf + buf * A_TILE, BsBuf + buf * B_TILE, tid);
          },
          AsBuf, BsBuf, acc, wr, wc, lane);

  const int g = lane >> 4, lm = lane & 15;
#pragma unroll
  for (int tm = 0; tm < 2; ++tm)
#pragma unroll
    for (int tn = 0; tn < 2; ++tn) {
      const int j = n0 + wc + tn * 16 + lm;
      const float bj = bo[j];
#pragma unroll
      for (int r = 0; r < 8; ++r) {
        const int m = m0 + wr + tm * 16 + g * 8 + r;
        out[(size_t)m * DIMX + j] = acc[tm][tn][r] + bj;
      }
    }
}

// ---------------------------------------------------------------------------
extern "C" void kernel_launch(void* const* d_in, const int* in_sizes, int n_in,
                              void* d_out, int out_size, void* d_ws,
                              size_t ws_size, hipStream_t stream)
{
  const float* x   = (const float*)d_in[0];
  const float* met = (const float*)d_in[1];
  const float* yz  = (const float*)d_in[2];
  const float* Wq  = (const float*)d_in[3];
  const float* Wk  = (const float*)d_in[4];
  const float* Wv  = (const float*)d_in[5];
  const float* Wo  = (const float*)d_in[6];
  const float* bo  = (const float*)d_in[7];

  float* out  = (float*)d_out;                              // (4,2048,512)
  float* attn = out + (size_t)BATCH * NSEQ * DIMX;          // (32,2048,2048)

  float* Qc  = (float*)d_ws;                                // 32*2048*128
  float* KcT = Qc + (size_t)BHT * NSEQ * DKC;               // 32*128*2048 (transposed)
  float* Vh  = KcT + (size_t)BHT * DKC * NSEQ;              // 32*2048*64
  float* Oh  = Vh + (size_t)BHT * NSEQ * DHEAD;             // 32*2048*64

  const dim3 blk(256);
  proj_kernel<<<dim3(DIMX / 64, (BATCH * NSEQ) / 128, 5), blk, 0, stream>>>(
      x, met, yz, Wq, Wk, Wv, Qc, KcT, Vh);
  qk_kernel<<<dim3(NSEQ / 64, NSEQ / 128, BHT), blk, 0, stream>>>(Qc, KcT, attn);
  softmax_kernel<<<dim3(BHT * NSEQ), blk, 0, stream>>>(attn);
  av_kernel<<<dim3(1, NSEQ / 128, BHT), blk, 0, stream>>>(attn, Vh, Oh);
  out_kernel<<<dim3(DIMX / 64, (BATCH * NSEQ) / 128), blk, 0, stream>>>(
      Oh, Wo, bo, out);
}